// HeteroGNN_52879637348658
// MI455X (gfx1250) — compile-verified
//
#include <hip/hip_runtime.h>

typedef __attribute__((ext_vector_type(2))) float v2f;
typedef __attribute__((ext_vector_type(8))) float v8f;

#define SROW 132   // LDS row stride (floats): 132 % 64 == 4 -> conflict-free b64 frags

static __device__ __forceinline__ int imin(int a, int b) { return a < b ? a : b; }

// ---------------------------------------------------------------- utilities
__global__ void hg_zero(float* __restrict__ p, long n) {
  long i = (long)blockIdx.x * blockDim.x + threadIdx.x;
  long stride = (long)gridDim.x * blockDim.x;
  for (; i < n; i += stride) p[i] = 0.0f;
}

__global__ void hg_count(const int* __restrict__ idx, float* __restrict__ cnt, int E) {
  int e = blockIdx.x * blockDim.x + threadIdx.x;
  if (e < E) atomicAdd(&cnt[idx[e]], 1.0f);
}

// acc[edst[e]][f] += x[esrc[e]][f]    (one thread per (edge, feature))
__global__ void hg_scatter(const float* __restrict__ x, const int* __restrict__ esrc,
                           const int* __restrict__ edst, float* __restrict__ acc,
                           int E, int logF) {
  long t = (long)blockIdx.x * blockDim.x + threadIdx.x;
  long total = (long)E << logF;
  if (t >= total) return;
  int F = 1 << logF;
  int e = (int)(t >> logF);
  int f = (int)(t & (F - 1));
  atomicAdd(&acc[(long)edst[e] * F + f], x[(long)esrc[e] * F + f]);
}

// acc[i][f] /= max(cnt[i], 1)
__global__ void hg_rownorm(float* __restrict__ acc, const float* __restrict__ cnt,
                           int N, int logF) {
  long t = (long)blockIdx.x * blockDim.x + threadIdx.x;
  if (t >= ((long)N << logF)) return;
  int i = (int)(t >> logF);
  acc[t] = acc[t] / fmaxf(cnt[i], 1.0f);
}

// ------------------------------------------------------------ weight repack
// Wp[(k4*64 + n)*4 + h*2 + j] = Wcat[k4*4 + h*2 + j][n]  where Wcat = [W1; W2]
// -> B fragment for WMMA k-step k4 is one contiguous 8-byte load per lane.
__global__ void hg_repack(const float* __restrict__ W1, const float* __restrict__ W2,
                          float* __restrict__ Wp, int F1, int F2) {
  int t = blockIdx.x * blockDim.x + threadIdx.x;
  int K = F1 + F2;
  if (t >= K * 64) return;
  int j  = t & 1;
  int h  = (t >> 1) & 1;
  int n  = (t >> 2) & 63;
  int k4 = t >> 8;
  int kr = k4 * 4 + h * 2 + j;
  Wp[t] = (kr < F1) ? W1[kr * 64 + n] : W2[(kr - F1) * 64 + n];
}

// ------------------------------------------------- fused SAGE linear (WMMA)
// out[N,64] = relu( [A1 | A2][N,K] @ Wp + bias ),  K = F1+F2 (64 or 128)
// 128 threads = 4 waves; block: 16 rows staged once into LDS; wave w: cols 16w..
__global__ void __launch_bounds__(128)
hg_sage_wmma(const float* __restrict__ A1, int F1,
             const float* __restrict__ A2, int F2,
             const float* __restrict__ Wp, const float* __restrict__ bias,
             float* __restrict__ out, int N, int logkq) {
  __shared__ float sA[16 * SROW];
  const int K  = F1 + F2;
  const int kq = 1 << logkq;                 // float4 slots per row = K/4
  const int r0 = blockIdx.x * 16;

  // cooperative stage of the 16 x K A-tile (rows clamped; EXEC stays full)
  for (int t = threadIdx.x; t < (16 << logkq); t += 128) {
    int i = t >> logkq;
    int q = t & (kq - 1);
    int row = imin(r0 + i, N - 1);
    int kk = q << 2;
    float4 v = (kk < F1) ? *(const float4*)&A1[(long)row * F1 + kk]
                         : *(const float4*)&A2[(long)row * F2 + (kk - F1)];
    *(float4*)&sA[i * SROW + kk] = v;
  }
  __syncthreads();

  const int lane = threadIdx.x & 31;
  const int wv = threadIdx.x >> 5;
  const int n0 = wv * 16;
  const int m  = lane & 15;          // A row / B,C col within tile
  const int hh = lane >> 4;          // lane half selects K pair (ISA A/B layout)
  const int kg = hh << 1;

  v8f c = {};
  const int K4 = K >> 2;
  for (int k4 = 0; k4 < K4; ++k4) {
    v2f a = *(const v2f*)&sA[m * SROW + (k4 << 2) + kg];
    v2f b = *(const v2f*)&Wp[((k4 << 6) + n0 + m) * 4 + (hh << 1)];
    c = __builtin_amdgcn_wmma_f32_16x16x4_f32(false, a, false, b, (short)0, c, false, false);
  }

  // C/D layout: VGPR r -> M = r + 8*half, N = lane&15
  const int cm = hh << 3;
  const int cn = m;
  const float bv = bias[n0 + cn];
#pragma unroll
  for (int r = 0; r < 8; ++r) {
    int orow = r0 + cm + r;
    if (orow < N) out[(long)orow * 64 + n0 + cn] = fmaxf(c[r] + bv, 0.0f);
  }
}

// ------------------------------------------------------- edge scorer (WMMA)
// logit[e] = relu(concat(acc2[src[e]], mer2[dst[e]]) @ w1 + b1) @ w2 + b2
// Gathered 16x128 edge-feature tile staged once into LDS, shared by 4 waves.
__global__ void __launch_bounds__(128)
hg_edge_mlp(const float* __restrict__ acc2, const float* __restrict__ mer2,
            const int* __restrict__ src, const int* __restrict__ dst,
            const float* __restrict__ Wp, const float* __restrict__ b1,
            const float* __restrict__ w2, const float* __restrict__ b2,
            float* __restrict__ logit, int E) {
  __shared__ float sE[16 * SROW];
  __shared__ float hs[16 * 64];
  const int e0 = blockIdx.x * 16;

  // stage: row i = concat(acc2[src[e_i]], mer2[dst[e_i]])  (32 float4 per row)
  for (int t = threadIdx.x; t < 16 * 32; t += 128) {
    int i  = t >> 5;
    int q  = t & 31;
    int e  = imin(e0 + i, E - 1);
    int kk = q << 2;
    const float* rp = (kk < 64) ? acc2 + (long)src[e] * 64 + kk
                                : mer2 + (long)dst[e] * 64 + (kk - 64);
    *(float4*)&sE[i * SROW + kk] = *(const float4*)rp;
  }
  __syncthreads();

  const int lane = threadIdx.x & 31;
  const int wv = threadIdx.x >> 5;
  const int n0 = wv * 16;
  const int m  = lane & 15;
  const int hh = lane >> 4;
  const int kg = hh << 1;

  v8f c = {};
#pragma unroll 4
  for (int k4 = 0; k4 < 32; ++k4) {      // K = 128
    v2f a = *(const v2f*)&sE[m * SROW + (k4 << 2) + kg];
    v2f b = *(const v2f*)&Wp[((k4 << 6) + n0 + m) * 4 + (hh << 1)];
    c = __builtin_amdgcn_wmma_f32_16x16x4_f32(false, a, false, b, (short)0, c, false, false);
  }

  const int cm = hh << 3;
  const int cn = m;
  const float bv = b1[n0 + cn];
#pragma unroll
  for (int r = 0; r < 8; ++r)
    hs[(cm + r) * 64 + n0 + cn] = fmaxf(c[r] + bv, 0.0f);
  __syncthreads();

  if (threadIdx.x < 16) {
    int ee = e0 + threadIdx.x;
    if (ee < E) {
      float s = b2[0];
#pragma unroll 8
      for (int j = 0; j < 64; ++j) s = fmaf(hs[threadIdx.x * 64 + j], w2[j], s);
      logit[ee] = s;
    }
  }
}

// ------------------------------------------------------------------ launch
extern "C" void kernel_launch(void* const* d_in, const int* in_sizes, int n_in,
                              void* d_out, int out_size, void* d_ws, size_t ws_size,
                              hipStream_t stream) {
  const int FIN = 32, H = 64;
  const float* x_acc   = (const float*)d_in[0];
  const float* x_mer   = (const float*)d_in[1];
  const int*   src     = (const int*)d_in[2];
  const int*   dst     = (const int*)d_in[3];
  const float* w1_am_l = (const float*)d_in[4];
  const float* b1_am_l = (const float*)d_in[5];
  const float* w1_am_r = (const float*)d_in[6];
  const float* w1_ma_l = (const float*)d_in[7];
  const float* b1_ma_l = (const float*)d_in[8];
  const float* w1_ma_r = (const float*)d_in[9];
  const float* w2_am_l = (const float*)d_in[10];
  const float* b2_am_l = (const float*)d_in[11];
  const float* w2_am_r = (const float*)d_in[12];
  const float* w2_ma_l = (const float*)d_in[13];
  const float* b2_ma_l = (const float*)d_in[14];
  const float* w2_ma_r = (const float*)d_in[15];
  const float* mlp_w1  = (const float*)d_in[16];
  const float* mlp_b1  = (const float*)d_in[17];
  const float* mlp_w2  = (const float*)d_in[18];
  const float* mlp_b2  = (const float*)d_in[19];
  float* out = (float*)d_out;

  const int NA = in_sizes[0] / FIN;
  const int NM = in_sizes[1] / FIN;
  const int E  = in_sizes[2];

  // workspace layout (aggM/aggA reused across layers 1 and 2)
  char* ws = (char*)d_ws;
  size_t off = 0;
  auto take = [&](size_t bytes) -> float* {
    float* p = (float*)(ws + off);
    off = (off + bytes + 255) & ~(size_t)255;
    return p;
  };
  float* cnt_m = take((size_t)NM * 4);
  float* cnt_a = take((size_t)NA * 4);
  float* aggM  = take((size_t)NM * H * 4);   // layer1 uses first NM*32
  float* aggA  = take((size_t)NA * H * 4);   // layer1 uses first NA*32
  float* mer1  = take((size_t)NM * H * 4);
  float* acc1  = take((size_t)NA * H * 4);
  float* mer2  = take((size_t)NM * H * 4);
  float* acc2  = take((size_t)NA * H * 4);
  float* wp1am = take((size_t)64  * 64 * 4);   // [w1_am_l ; w1_am_r] repacked
  float* wp1ma = take((size_t)64  * 64 * 4);
  float* wp2am = take((size_t)128 * 64 * 4);
  float* wp2ma = take((size_t)128 * 64 * 4);
  float* wpmlp = take((size_t)128 * 64 * 4);

  const int TB = 256;
  auto zero = [&](float* p, long n) {
    int blocks = (int)((n + TB - 1) / TB);
    if (blocks > 4096) blocks = 4096;
    hipLaunchKernelGGL(hg_zero, dim3(blocks), dim3(TB), 0, stream, p, n);
  };
  auto repack = [&](const float* W1, const float* W2, float* Wp, int F1, int F2) {
    int n = (F1 + F2) * 64;
    hipLaunchKernelGGL(hg_repack, dim3((n + TB - 1) / TB), dim3(TB), 0, stream,
                       W1, W2, Wp, F1, F2);
  };

  // weight repacks (tiny, stay hot in L2 for every GEMM block)
  repack(w1_am_l, w1_am_r, wp1am, FIN, FIN);
  repack(w1_ma_l, w1_ma_r, wp1ma, FIN, FIN);
  repack(w2_am_l, w2_am_r, wp2am, H, H);
  repack(w2_ma_l, w2_ma_r, wp2ma, H, H);
  repack(mlp_w1,  mlp_w1,  wpmlp, 2 * H, 0);

  // edge counts (same graph for both layers)
  zero(cnt_m, NM);
  zero(cnt_a, NA);
  hipLaunchKernelGGL(hg_count, dim3((E + TB - 1) / TB), dim3(TB), 0, stream, dst, cnt_m, E);
  hipLaunchKernelGGL(hg_count, dim3((E + TB - 1) / TB), dim3(TB), 0, stream, src, cnt_a, E);

  // ---- layer 1: account -> merchant
  zero(aggM, (long)NM * FIN);
  hipLaunchKernelGGL(hg_scatter, dim3((int)(((long)E * FIN + TB - 1) / TB)), dim3(TB), 0, stream,
                     x_acc, src, dst, aggM, E, 5);
  hipLaunchKernelGGL(hg_rownorm, dim3((int)(((long)NM * FIN + TB - 1) / TB)), dim3(TB), 0, stream,
                     aggM, cnt_m, NM, 5);
  hipLaunchKernelGGL(hg_sage_wmma, dim3((NM + 15) / 16), dim3(128), 0, stream,
                     aggM, FIN, x_mer, FIN, wp1am, b1_am_l, mer1, NM, 4);

  // ---- layer 1: merchant -> account
  zero(aggA, (long)NA * FIN);
  hipLaunchKernelGGL(hg_scatter, dim3((int)(((long)E * FIN + TB - 1) / TB)), dim3(TB), 0, stream,
                     x_mer, dst, src, aggA, E, 5);
  hipLaunchKernelGGL(hg_rownorm, dim3((int)(((long)NA * FIN + TB - 1) / TB)), dim3(TB), 0, stream,
                     aggA, cnt_a, NA, 5);
  hipLaunchKernelGGL(hg_sage_wmma, dim3((NA + 15) / 16), dim3(128), 0, stream,
                     aggA, FIN, x_acc, FIN, wp1ma, b1_ma_l, acc1, NA, 4);

  // ---- layer 2: account -> merchant (acc1 features)
  zero(aggM, (long)NM * H);
  hipLaunchKernelGGL(hg_scatter, dim3((int)(((long)E * H + TB - 1) / TB)), dim3(TB), 0, stream,
                     acc1, src, dst, aggM, E, 6);
  hipLaunchKernelGGL(hg_rownorm, dim3((int)(((long)NM * H + TB - 1) / TB)), dim3(TB), 0, stream,
                     aggM, cnt_m, NM, 6);
  hipLaunchKernelGGL(hg_sage_wmma, dim3((NM + 15) / 16), dim3(128), 0, stream,
                     aggM, H, mer1, H, wp2am, b2_am_l, mer2, NM, 5);

  // ---- layer 2: merchant -> account (mer1 features)
  zero(aggA, (long)NA * H);
  hipLaunchKernelGGL(hg_scatter, dim3((int)(((long)E * H + TB - 1) / TB)), dim3(TB), 0, stream,
                     mer1, dst, src, aggA, E, 6);
  hipLaunchKernelGGL(hg_rownorm, dim3((int)(((long)NA * H + TB - 1) / TB)), dim3(TB), 0, stream,
                     aggA, cnt_a, NA, 6);
  hipLaunchKernelGGL(hg_sage_wmma, dim3((NA + 15) / 16), dim3(128), 0, stream,
                     aggA, H, acc1, H, wp2ma, b2_ma_l, acc2, NA, 5);

  // ---- edge scorer
  hipLaunchKernelGGL(hg_edge_mlp, dim3((E + 15) / 16), dim3(128), 0, stream,
                     acc2, mer2, src, dst, wpmlp, mlp_b1, mlp_w2, mlp_b2, out, E);

  (void)n_in; (void)out_size; (void)ws_size;
}